// Generator_31370441130469
// MI455X (gfx1250) — compile-verified
//
#include <hip/hip_runtime.h>
#include <hip/hip_bf16.h>
#include <math.h>

typedef __attribute__((ext_vector_type(2))) float v2f;
typedef __attribute__((ext_vector_type(8))) float v8f;
typedef __attribute__((ext_vector_type(4))) int   v4i;

#define C_CH   128
#define L_LEN  4096
#define B_N    8
#define P_OUT  96
#define D_ATT  16
#define BN_EPS 1e-5f

__device__ __forceinline__ v8f wmma4(v2f a, v2f b, v8f c) {
  // D(16x16,f32) = A(16x4,f32) * B(4x16,f32) + C
  return __builtin_amdgcn_wmma_f32_16x16x4_f32(false, a, false, b, (short)0, c, false, false);
}

// ---- async global->LDS path (gfx1250), guarded so compile cannot regress ----
#if __has_builtin(__builtin_amdgcn_global_load_async_to_lds_b128) && \
    __has_builtin(__builtin_amdgcn_global_load_async_to_lds_b32)  && \
    __has_builtin(__builtin_amdgcn_s_wait_asynccnt)
#define USE_ASYNC_LDS 1
#else
#define USE_ASYNC_LDS 0
#endif

#if USE_ASYNC_LDS
typedef __attribute__((address_space(1))) int  g_i32;
typedef __attribute__((address_space(3))) int  l_i32;
typedef __attribute__((address_space(1))) v4i  g_v4i;
typedef __attribute__((address_space(3))) v4i  l_v4i;

__device__ __forceinline__ void async_g2l_b32(const void* g, void* l) {
  __builtin_amdgcn_global_load_async_to_lds_b32(
      (g_i32*)(void*)g, (l_i32*)l, 0, 0);
}
__device__ __forceinline__ void async_g2l_b128(const void* g, void* l) {
  __builtin_amdgcn_global_load_async_to_lds_b128(
      (g_v4i*)(void*)g, (l_v4i*)l, 0, 0);
}
#endif

// ---------------------------------------------------------------------------
// K1: h[b,o,l] = sum_i x[b,i] * Wt[i,o,l] + bt[o]   (streams 256MB of Wt)
//     plus deterministic per-block BN partial sums (sum, sumsq) per channel.
// grid (L/256, C), block 256
// ---------------------------------------------------------------------------
__global__ __launch_bounds__(256) void convt_bn_stats(
    const float* __restrict__ x, const float* __restrict__ Wt,
    const float* __restrict__ bt, float* __restrict__ h,
    float* __restrict__ part1, float* __restrict__ part2) {
  __shared__ float xs[B_N * C_CH];
  __shared__ float red1[256];
  __shared__ float red2[256];
  const int tid = threadIdx.x;
  for (int i = tid; i < B_N * C_CH; i += 256) xs[i] = x[i];
  __syncthreads();

  const int o = blockIdx.y;
  const int l = blockIdx.x * 256 + tid;
  float acc[B_N];
#pragma unroll
  for (int b = 0; b < B_N; b++) acc[b] = 0.f;

  const size_t istride = (size_t)C_CH * L_LEN;
  const float* wp = Wt + (size_t)o * L_LEN + l;
  for (int i = 0; i < C_CH; i++) {
    if (i + 4 < C_CH) __builtin_prefetch(wp + (size_t)(i + 4) * istride, 0, 0);
    const float w = wp[(size_t)i * istride];
#pragma unroll
    for (int b = 0; b < B_N; b++) acc[b] = fmaf(xs[b * C_CH + i], w, acc[b]);
  }
  const float bo = bt[o];
  float s1 = 0.f, s2 = 0.f;
#pragma unroll
  for (int b = 0; b < B_N; b++) {
    const float hv = acc[b] + bo;
    h[((size_t)b * C_CH + o) * L_LEN + l] = hv;
    s1 += hv;
    s2 += hv * hv;
  }
  red1[tid] = s1;
  red2[tid] = s2;
  __syncthreads();
  for (int s = 128; s > 0; s >>= 1) {
    if (tid < s) { red1[tid] += red1[tid + s]; red2[tid] += red2[tid + s]; }
    __syncthreads();
  }
  if (tid == 0) {
    part1[o * gridDim.x + blockIdx.x] = red1[0];
    part2[o * gridDim.x + blockIdx.x] = red2[0];
  }
}

// ---------------------------------------------------------------------------
// K2: finalize BN -> per-channel scale/shift.  1 block x 128 threads
// ---------------------------------------------------------------------------
__global__ void bn_finalize(const float* __restrict__ part1,
                            const float* __restrict__ part2,
                            const float* __restrict__ gamma,
                            const float* __restrict__ beta,
                            float* __restrict__ scale, float* __restrict__ shift) {
  const int o = threadIdx.x;
  float s1 = 0.f, s2 = 0.f;
  for (int j = 0; j < 16; j++) { s1 += part1[o * 16 + j]; s2 += part2[o * 16 + j]; }
  const float inv = 1.0f / (float)(B_N * L_LEN);
  const float mean = s1 * inv;
  const float var = s2 * inv - mean * mean;
  const float sc = gamma[o] * rsqrtf(var + BN_EPS);
  scale[o] = sc;
  shift[o] = beta[o] - mean * sc;
}

// ---------------------------------------------------------------------------
// K3: hn = relu(bn(h)); q = Wq*hn+bq, k = Wk*hn+bk, v = Wv*hn+bv  (WMMA f32)
// One wave per 16x16 output tile. 10 row-tiles (q:1, k:1, v:8) x 256 l-tiles x 8 b.
// v stored transposed as [b][l][c] so the attention kernel loads coalesced.
// grid 2560, block 256 (8 waves)
// ---------------------------------------------------------------------------
__global__ __launch_bounds__(256) void qkv_proj(
    const float* __restrict__ h, const float* __restrict__ scale,
    const float* __restrict__ shift,
    const float* __restrict__ Wq, const float* __restrict__ bq,
    const float* __restrict__ Wk, const float* __restrict__ bk,
    const float* __restrict__ Wv, const float* __restrict__ bv,
    float* __restrict__ q, float* __restrict__ k, float* __restrict__ v) {
  const int wid  = blockIdx.x * 8 + (threadIdx.x >> 5);
  const int lane = threadIdx.x & 31;
  const int col  = lane & 15;
  const int hi   = lane >> 4;
  const int lt = wid & 255;
  const int rt = (wid >> 8) % 10;
  const int b  = wid / (256 * 10);
  const int l0 = lt * 16;

  const float* W; const float* bias; int row0;
  if (rt == 0)      { W = Wq; bias = bq; row0 = 0; }
  else if (rt == 1) { W = Wk; bias = bk; row0 = 0; }
  else              { W = Wv; bias = bv; row0 = (rt - 2) * 16; }

  v8f acc = {};
  const int m = row0 + col;
#pragma unroll 4
  for (int kk = 0; kk < C_CH; kk += 4) {
    const int c0 = kk + 2 * hi;
    v2f a, bb;
    a.x = W[m * C_CH + c0];
    a.y = W[m * C_CH + c0 + 1];
    const float h0 = h[((size_t)b * C_CH + c0) * L_LEN + l0 + col];
    const float h1 = h[((size_t)b * C_CH + c0 + 1) * L_LEN + l0 + col];
    bb.x = fmaxf(fmaf(h0, scale[c0], shift[c0]), 0.f);
    bb.y = fmaxf(fmaf(h1, scale[c0 + 1], shift[c0 + 1]), 0.f);
    acc = wmma4(a, bb, acc);
  }
#pragma unroll
  for (int r = 0; r < 8; r++) {
    const int row = row0 + r + 8 * hi;
    const float val = acc[r] + bias[row];
    if (rt == 0)      q[((size_t)b * D_ATT + row) * L_LEN + l0 + col] = val;
    else if (rt == 1) k[((size_t)b * D_ATT + row) * L_LEN + l0 + col] = val;
    else              v[((size_t)b * L_LEN + l0 + col) * C_CH + row] = val; // [b][l][c]
  }
}

// ---------------------------------------------------------------------------
// K4: flash attention.  energy = q^T k (K=d=16), online softmax, out += V P^T.
// One wave owns 16 l-rows and all 128 channels (8 accum tiles).
// Block = 8 waves share the V/K m-panels via LDS.  Panels are double-buffered
// and filled with async global->LDS DMA (ASYNCcnt) so the DMA of tile mt+1
// overlaps the WMMA/softmax work on tile mt.
// grid 256 (= 8 b x 32 l-blocks), block 256
// ---------------------------------------------------------------------------
__global__ __launch_bounds__(256) void attn_fused(
    const float* __restrict__ q, const float* __restrict__ k,
    const float* __restrict__ v, float* __restrict__ outb) {
  __shared__ __align__(16) float v_lds[2][16][C_CH]; // [buf][m_local][c]
  __shared__ __align__(16) float k_lds[2][16][16];   // [buf][d][m_local]
  __shared__ float p_lds[8][16][17];  // per-wave P tile [l_local][m_local], padded
  __shared__ float st_lds[8][16];     // per-wave per-row softmax factors

  const int tid  = threadIdx.x;
  const int wave = tid >> 5;
  const int lane = tid & 31;
  const int col  = lane & 15;
  const int hi   = lane >> 4;
  const int b    = blockIdx.x >> 5;
  const int lb   = blockIdx.x & 31;
  const int l0   = (lb * 8 + wave) * 16;
  const int NT   = L_LEN / 16;

#if USE_ASYNC_LDS
  // panel DMA: k panel 16x16 (one b32/lane), v panel 16x128 (two b128/lane)
  const int kd  = tid >> 4, kmm = tid & 15;
  const int vmm = tid >> 5, vc4 = (tid & 31) * 4;
#define ISSUE_PANELS(BUF, M0)                                                 \
  do {                                                                        \
    async_g2l_b32(&k[((size_t)b * D_ATT + kd) * L_LEN + (M0) + kmm],          \
                  &k_lds[BUF][kd][kmm]);                                      \
    async_g2l_b128(&v[((size_t)b * L_LEN + (M0) + vmm) * C_CH + vc4],         \
                   &v_lds[BUF][vmm][vc4]);                                    \
    async_g2l_b128(&v[((size_t)b * L_LEN + (M0) + vmm + 8) * C_CH + vc4],     \
                   &v_lds[BUF][vmm + 8][vc4]);                                \
  } while (0)
  ISSUE_PANELS(0, 0);
#endif

  // A operand for energy: q^T tile (rows = l, K = d), resident for whole sweep
  v2f aQ[4];
#pragma unroll
  for (int kc = 0; kc < 4; kc++) {
    const int d0 = kc * 4 + 2 * hi;
    aQ[kc].x = q[((size_t)b * D_ATT + d0) * L_LEN + l0 + col];
    aQ[kc].y = q[((size_t)b * D_ATT + d0 + 1) * L_LEN + l0 + col];
  }

  const v8f vzero = {};
  v8f acc[8];
#pragma unroll
  for (int ct = 0; ct < 8; ct++) acc[ct] = vzero;
  float mrun[8], lrun[8];
#pragma unroll
  for (int r = 0; r < 8; r++) { mrun[r] = -3.0e38f; lrun[r] = 0.f; }

  for (int mt = 0; mt < NT; mt++) {
    const int buf = mt & 1;
#if USE_ASYNC_LDS
    __builtin_amdgcn_s_wait_asynccnt(0);  // my panel DMA for tile mt landed
    __syncthreads();                      // publish panel; everyone left buf^1
    if (mt + 1 < NT) ISSUE_PANELS(buf ^ 1, (mt + 1) * 16);  // overlaps compute
#else
    const int m0s = mt * 16;
    __syncthreads();
    {
      const int d = tid >> 4, mm = tid & 15;
      k_lds[buf][d][mm] = k[((size_t)b * D_ATT + d) * L_LEN + m0s + mm];
    }
#pragma unroll
    for (int j = 0; j < 8; j++) {
      const int idx = j * 256 + tid;
      const int mm = idx >> 7, cc = idx & 127;
      v_lds[buf][mm][cc] = v[((size_t)b * L_LEN + m0s + mm) * C_CH + cc];
    }
    __syncthreads();
#endif

    // energy tile E[16l x 16m]
    v8f e = vzero;
#pragma unroll
    for (int kc = 0; kc < 4; kc++) {
      const int d0 = kc * 4 + 2 * hi;
      v2f bk_;
      bk_.x = k_lds[buf][d0][col];
      bk_.y = k_lds[buf][d0 + 1][col];
      e = wmma4(aQ[kc], bk_, e);
    }

    // online softmax per row (16-lane group reductions)
#pragma unroll
    for (int r = 0; r < 8; r++) {
      float rmax = e[r];
      rmax = fmaxf(rmax, __shfl_xor(rmax, 1, 32));
      rmax = fmaxf(rmax, __shfl_xor(rmax, 2, 32));
      rmax = fmaxf(rmax, __shfl_xor(rmax, 4, 32));
      rmax = fmaxf(rmax, __shfl_xor(rmax, 8, 32));
      const float newm = fmaxf(mrun[r], rmax);
      const float pr = __expf(e[r] - newm);
      float rs = pr;
      rs += __shfl_xor(rs, 1, 32);
      rs += __shfl_xor(rs, 2, 32);
      rs += __shfl_xor(rs, 4, 32);
      rs += __shfl_xor(rs, 8, 32);
      const float cf = __expf(mrun[r] - newm);
      lrun[r] = lrun[r] * cf + rs;
      mrun[r] = newm;
      p_lds[wave][r + 8 * hi][col] = pr;
      if (col == 0) st_lds[wave][r + 8 * hi] = cf;
    }

    // rescale accumulators: factor indexed by output column l
    const float f = st_lds[wave][col];
#pragma unroll
    for (int ct = 0; ct < 8; ct++) acc[ct] *= f;

    // out[c,l] += V(c x m) * P^T(m x l)
#pragma unroll
    for (int ct = 0; ct < 8; ct++) {
#pragma unroll
      for (int kc = 0; kc < 4; kc++) {
        const int mm = kc * 4 + 2 * hi;
        v2f av, bp;
        av.x = v_lds[buf][mm][ct * 16 + col];
        av.y = v_lds[buf][mm + 1][ct * 16 + col];
        bp.x = p_lds[wave][col][mm];
        bp.y = p_lds[wave][col][mm + 1];
        acc[ct] = wmma4(av, bp, acc[ct]);
      }
    }
  }

  // normalize by row sums (per column l) and store out [b][c][L]
#pragma unroll
  for (int r = 0; r < 8; r++)
    if (col == 0) st_lds[wave][r + 8 * hi] = lrun[r];
  const float linv = 1.0f / st_lds[wave][col];
#pragma unroll
  for (int ct = 0; ct < 8; ct++) {
#pragma unroll
    for (int r = 0; r < 8; r++) {
      const int cch = ct * 16 + r + 8 * hi;
      outb[((size_t)b * C_CH + cch) * L_LEN + l0 + col] = acc[ct][r] * linv;
    }
  }
}

// ---------------------------------------------------------------------------
// K5: y = W2 * out + b2  (96x128 GEMM via WMMA).  grid 1536, block 256
// ---------------------------------------------------------------------------
__global__ __launch_bounds__(256) void final_proj(
    const float* __restrict__ ob, const float* __restrict__ W2,
    const float* __restrict__ b2, float* __restrict__ y) {
  const int wid  = blockIdx.x * 8 + (threadIdx.x >> 5);
  const int lane = threadIdx.x & 31;
  const int col  = lane & 15;
  const int hi   = lane >> 4;
  const int lt = wid & 255;
  const int rt = (wid >> 8) % 6;
  const int b  = wid / (256 * 6);
  const int l0 = lt * 16;
  const int row0 = rt * 16;

  v8f acc = {};
  const int m = row0 + col;
#pragma unroll 4
  for (int kk = 0; kk < C_CH; kk += 4) {
    const int c0 = kk + 2 * hi;
    v2f a, bb;
    a.x = W2[m * C_CH + c0];
    a.y = W2[m * C_CH + c0 + 1];
    bb.x = ob[((size_t)b * C_CH + c0) * L_LEN + l0 + col];
    bb.y = ob[((size_t)b * C_CH + c0 + 1) * L_LEN + l0 + col];
    acc = wmma4(a, bb, acc);
  }
#pragma unroll
  for (int r = 0; r < 8; r++) {
    const int row = row0 + r + 8 * hi;
    y[((size_t)b * P_OUT + row) * L_LEN + l0 + col] = acc[r] + b2[row];
  }
}

// ---------------------------------------------------------------------------
extern "C" void kernel_launch(void* const* d_in, const int* in_sizes, int n_in,
                              void* d_out, int out_size, void* d_ws, size_t ws_size,
                              hipStream_t stream) {
  const float* x     = (const float*)d_in[0];
  const float* Wt    = (const float*)d_in[1];
  const float* bt    = (const float*)d_in[2];
  const float* gamma = (const float*)d_in[3];
  const float* beta  = (const float*)d_in[4];
  const float* Wq    = (const float*)d_in[5];
  const float* bq    = (const float*)d_in[6];
  const float* Wk    = (const float*)d_in[7];
  const float* bk    = (const float*)d_in[8];
  const float* Wv    = (const float*)d_in[9];
  const float* bv    = (const float*)d_in[10];
  const float* W2    = (const float*)d_in[11];
  const float* b2    = (const float*)d_in[12];
  float* y = (float*)d_out;

  float* ws    = (float*)d_ws;
  float* h     = ws;                                   // B*C*L, reused as attn out
  float* part1 = h + (size_t)B_N * C_CH * L_LEN;       // 2048
  float* part2 = part1 + 2048;                         // 2048
  float* scale = part2 + 2048;                         // 128
  float* shift = scale + 128;                          // 128
  float* q     = shift + 128;                          // B*16*L
  float* k     = q + (size_t)B_N * D_ATT * L_LEN;      // B*16*L
  float* v     = k + (size_t)B_N * D_ATT * L_LEN;      // B*L*C  ([b][l][c])
  float* ob    = h;                                    // reuse h after K3

  convt_bn_stats<<<dim3(L_LEN / 256, C_CH), 256, 0, stream>>>(x, Wt, bt, h, part1, part2);
  bn_finalize<<<1, C_CH, 0, stream>>>(part1, part2, gamma, beta, scale, shift);
  qkv_proj<<<(B_N * 10 * 256) / 8, 256, 0, stream>>>(h, scale, shift, Wq, bq, Wk, bk,
                                                     Wv, bv, q, k, v);
  attn_fused<<<B_N * (L_LEN / 128), 256, 0, stream>>>(q, k, v, ob);
  final_proj<<<(B_N * 6 * 256) / 8, 256, 0, stream>>>(ob, W2, b2, y);
}